// VQC_77876347011708
// MI455X (gfx1250) — compile-verified
//
#include <hip/hip_runtime.h>

#define NQ      12
#define DIM     4096      // 2^12 amplitudes
#define BATCH   1024
#define LAYERS  3         // ANSATZ_COPIES + 1
#define THREADS 256       // 8 wave32 waves per workgroup

typedef __attribute__((ext_vector_type(2))) float v2f;
typedef __attribute__((ext_vector_type(8))) float v8f;

__global__ __launch_bounds__(THREADS)
void vqc_kernel(const float* __restrict__ inputs,   // [BATCH, NQ]
                const float* __restrict__ thetas,   // [LAYERS, NQ, 2] flat (72)
                float* __restrict__ out)            // [BATCH, NQ]
{
    __shared__ float re[DIM];
    __shared__ float im[DIM];
    __shared__ float partial[THREADS];   // per-lane WMMA partials

    const int b = blockIdx.x;
    const int t = threadIdx.x;

    const float INVSQRT2 = 0.70710678118654752440f;
    const float PI_F     = 3.14159265358979323846f;

    // ---- init |0...0> ----
    for (int k = t; k < DIM; k += THREADS) { re[k] = 0.0f; im[k] = 0.0f; }
    if (t == 0) re[0] = 1.0f;
    __syncthreads();

    // ---- encoding: per qubit q, fused (H then Z^{x_q}) ----
    // qubit q <-> flat-index bit (11 - q); pair stride = 1 << (11-q)
    for (int q = 0; q < NQ; ++q) {
        float x = inputs[b * NQ + q];           // block-uniform -> scalar load
        float ph_s, ph_c;
        __sincosf(PI_F * x, &ph_s, &ph_c);      // e^{i*pi*x} = (ph_c, ph_s)
        const int stride = 1 << (NQ - 1 - q);
        #pragma unroll
        for (int p = t; p < DIM / 2; p += THREADS) {
            int i0 = ((p & ~(stride - 1)) << 1) | (p & (stride - 1));
            int i1 = i0 | stride;
            float a0r = re[i0], a0i = im[i0];
            float a1r = re[i1], a1i = im[i1];
            float sr = INVSQRT2 * (a0r + a1r), si = INVSQRT2 * (a0i + a1i);
            float dr = INVSQRT2 * (a0r - a1r), di = INVSQRT2 * (a0i - a1i);
            re[i0] = sr;                      im[i0] = si;
            re[i1] = ph_c * dr - ph_s * di;   im[i1] = ph_c * di + ph_s * dr;
        }
        __syncthreads();
    }

    // ---- ansatz layers: fused (Y^t0 then Z^t1) per qubit, then CNOT-chain perm ----
    for (int l = 0; l < LAYERS; ++l) {
        for (int q = 0; q < NQ; ++q) {
            float t0 = thetas[(l * NQ + q) * 2 + 0];
            float t1 = thetas[(l * NQ + q) * 2 + 1];
            float rs, rc, ph_s, ph_c;
            __sincosf(0.5f * PI_F * t0, &rs, &rc);   // Ry: [[c,-s],[s,c]]
            __sincosf(PI_F * t1, &ph_s, &ph_c);      // Z^t1 phase on |1>
            const int stride = 1 << (NQ - 1 - q);
            #pragma unroll
            for (int p = t; p < DIM / 2; p += THREADS) {
                int i0 = ((p & ~(stride - 1)) << 1) | (p & (stride - 1));
                int i1 = i0 | stride;
                float a0r = re[i0], a0i = im[i0];
                float a1r = re[i1], a1i = im[i1];
                float n0r = rc * a0r - rs * a1r, n0i = rc * a0i - rs * a1i;
                float n1r = rs * a0r + rc * a1r, n1i = rs * a0i + rc * a1i;
                re[i0] = n0r;                     im[i0] = n0i;
                re[i1] = ph_c * n1r - ph_s * n1i; im[i1] = ph_c * n1i + ph_s * n1r;
            }
            __syncthreads();
        }
        if (l < LAYERS - 1) {
            // CNOT chain q->q+1 composed = one permutation:
            // new[y] = old[x], y = MSB-downward prefix-XOR of x's bits.
            float rr[16], ii[16];
            #pragma unroll
            for (int k = 0; k < 16; ++k) {
                int x = k * THREADS + t;
                rr[k] = re[x]; ii[k] = im[x];
            }
            __syncthreads();
            #pragma unroll
            for (int k = 0; k < 16; ++k) {
                int x = k * THREADS + t;
                int y = x; y ^= y >> 1; y ^= y >> 2; y ^= y >> 4; y ^= y >> 8;
                re[y] = rr[k]; im[y] = ii[k];
            }
            __syncthreads();
        }
    }

    // ---- probabilities in-place into re[] ----
    for (int k = t; k < DIM; k += THREADS) {
        float r = re[k], i2 = im[k];
        re[k] = r * r + i2 * i2;
    }
    __syncthreads();

    // ---- readout GEMM via V_WMMA_F32_16X16X4_F32 ----
    // exps[q] = sum_hi s_hi(hi,q) * ( sum_lo probs[hi*64+lo] * s_lo(lo,q) )
    // Stage 1: T[hi,q] = P[hi,lo] x S_lo[lo,q]  (M=64 -> 4 tiles, K=64 -> 16 chunks, N=16)
    // 8 waves: tile = wave&3, K-chunk half = wave>>2  (all waves active -> EXEC all-ones)
    const int wave = t >> 5;
    const int lane = t & 31;
    const int tile = wave & 3;
    const int cbase = (wave >> 2) * 8;

    const int m     = lane & 15;            // A row within tile / B column (qubit q)
    const int hi    = tile * 16 + m;
    const int kbase = (lane < 16) ? 0 : 2;  // lane-half K selector (A and B rows)
    const int q     = m;

    // qubit q's sign bit: position 11-q. lo covers bits 0..5 (qubits 6..11),
    // hi covers bits 6..11 (qubits 0..5).
    const bool lowq = (q >= 6) && (q < 12);
    const int  pb   = (11 - q) & 31;

    v8f acc = {};
    #pragma unroll
    for (int cc = 0; cc < 8; ++cc) {
        int c   = cbase + cc;
        int lo0 = c * 4 + kbase;
        int lo1 = lo0 + 1;
        v2f A, B;
        A.x = re[hi * 64 + lo0];
        A.y = re[hi * 64 + lo1];
        B.x = (lowq && ((lo0 >> pb) & 1)) ? -1.0f : 1.0f;
        B.y = (lowq && ((lo1 >> pb) & 1)) ? -1.0f : 1.0f;
        acc = __builtin_amdgcn_wmma_f32_16x16x4_f32(
                  /*neg_a=*/false, A, /*neg_b=*/false, B,
                  /*c_mod=*/(short)0, acc, /*reuse_a=*/false, /*reuse_b=*/false);
    }

    // Stage 2: apply hi-signs and fold the 8 C rows this lane owns.
    // C layout: VGPR r, lane -> M = r + (lane<16 ? 0 : 8), N = lane&15.
    const bool hiq = (q < 6);
    const int  pbh = (5 - q) & 31;
    float psum = 0.0f;
    #pragma unroll
    for (int r = 0; r < 8; ++r) {
        int M     = r + ((lane < 16) ? 0 : 8);
        int hiIdx = tile * 16 + M;
        float sgn = (hiq && ((hiIdx >> pbh) & 1)) ? -1.0f : 1.0f;
        psum += sgn * acc[r];
    }
    partial[wave * 32 + lane] = psum;
    __syncthreads();

    // ---- deterministic fixed-order cross-wave reduction, write output ----
    if (t < NQ) {
        float e = 0.0f;
        #pragma unroll
        for (int w = 0; w < 8; ++w)
            e += partial[w * 32 + t] + partial[w * 32 + 16 + t];
        out[b * NQ + t] = e;
    }
}

extern "C" void kernel_launch(void* const* d_in, const int* in_sizes, int n_in,
                              void* d_out, int out_size, void* d_ws, size_t ws_size,
                              hipStream_t stream) {
    const float* inputs = (const float*)d_in[0];   // [1024,12] f32
    const float* thetas = (const float*)d_in[1];   // [72] f32
    float* out = (float*)d_out;                    // [1024,12] f32
    vqc_kernel<<<dim3(BATCH), dim3(THREADS), 0, stream>>>(inputs, thetas, out);
}